// TTFHead_26268019982821
// MI455X (gfx1250) — compile-verified
//
#include <hip/hip_runtime.h>
#include <stdint.h>

// ---------------------------------------------------------------------------
// TTFHead decode for MI455X (gfx1250, wave32).
//   pred_hm [8,80,256,256] f32, pred_wh [8,4,256,256] f32 -> det [8,100,6] f32
// Memory-bound: ~176MB read once => ~7.6us floor @ 23.3 TB/s.
// CDNA5 features used: 320KB LDS/WGP (whole 256KB class plane resident),
// async global->LDS staging (global_load_async_to_lds_b128 + s_wait_asynccnt).
// Monotone sigmoid => rank on raw logits; sigmoid only on final winners.
// ---------------------------------------------------------------------------

#define BB   8
#define CC   80
#define HH   256
#define WW   256
#define HW   65536
#define TOPK 100
#define THR  0.02f

#if defined(__has_builtin)
#  if __has_builtin(__builtin_amdgcn_global_load_async_to_lds_b128)
#    define HAVE_ASYNC_COPY 1
#  endif
#  if __has_builtin(__builtin_amdgcn_s_wait_asynccnt)
#    define HAVE_WAIT_ASYNC 1
#  endif
#endif

#define AS1 __attribute__((address_space(1)))
#define AS3 __attribute__((address_space(3)))

typedef int v4i __attribute__((ext_vector_type(4)));

__device__ __forceinline__ AS1 v4i* to_as1_v4(const void* p) {
  // global flat address == AS1 address (numeric identity)
  return (AS1 v4i*)(unsigned long long)(uintptr_t)p;
}
__device__ __forceinline__ AS3 v4i* to_as3_v4(void* p) {
  // generic LDS pointer: low 32 bits are the LDS offset; AS3 ptrs are 32-bit
  return (AS3 v4i*)(unsigned)(uintptr_t)p;
}

__device__ __forceinline__ void async_copy16(const void* g, void* l) {
#if defined(HAVE_ASYNC_COPY)
  __builtin_amdgcn_global_load_async_to_lds_b128(to_as1_v4(g), to_as3_v4(l), 0, 0);
#else
  *(float4*)l = *(const float4*)g;
#endif
}
__device__ __forceinline__ void async_wait_all() {
#if defined(HAVE_ASYNC_COPY)
#  if defined(HAVE_WAIT_ASYNC)
  __builtin_amdgcn_s_wait_asynccnt(0);
#  else
  asm volatile("s_wait_asynccnt 0" ::: "memory");
#  endif
#endif
}

// monotone float -> orderable uint (and inverse). key 0 is reserved "invalid".
__device__ __forceinline__ unsigned fkey(float v) {
  unsigned b = __float_as_uint(v);
  return (b & 0x80000000u) ? ~b : (b | 0x80000000u);
}
__device__ __forceinline__ float finv(unsigned k) {
  unsigned b = (k & 0x80000000u) ? (k & 0x7fffffffu) : ~k;
  return __uint_as_float(b);  // finv(0) = NaN -> thresholded to zero row
}

// ---------------------------------------------------------------------------
// Kernel 1: one block per (b,c). Stage plane to LDS, sliding-window 3x3 peak
// mask in registers, 4-pass radix select of exact top-100 keys, emit packed
// (key,cls,spatial) u64 records. LDS: 256KB plane + 1KB hist + ctrl.
// ---------------------------------------------------------------------------
__global__ __launch_bounds__(1024) void k_class_topk(
    const float* __restrict__ hm, unsigned long long* __restrict__ cand) {
  const int blk = blockIdx.x;
  const int b = blk / CC, c = blk % CC;
  const int tid = threadIdx.x;

  extern __shared__ char smem[];
  float*    plane = (float*)smem;                    // 65536 f32
  unsigned* hist  = (unsigned*)(smem + HW * 4);      // 256
  unsigned* ctrl  = hist + 256;                      // [0]=pre [1]=msk [2]=k [3]=gt [4]=eq

  const float* src = hm + (size_t)(b * CC + c) * HW;

  // ---- stage whole class plane into LDS (one HBM pass, async DMA) ----
  for (int i = tid; i < HW / 4; i += 1024)
    async_copy16((const float4*)src + i, (float4*)plane + i);
  async_wait_all();
  __syncthreads();

  // ---- per-thread peak bitmask over its 64 consecutive elements ----
  // Clamped-index neighbors are exact-equivalent to -inf SAME padding:
  // a clamped neighbor duplicates an in-window comparison and can never
  // flip the "no neighbor strictly greater" decision.
  const int base = tid * 64;        // one quarter-row: same y for all 64
  const int y0 = base >> 8;
  const int x0 = base & 255;
  unsigned pb0 = 0u, pb1 = 0u;
  {
    const float* rm = plane + (y0 > 0 ? y0 - 1 : 0) * 256;
    const float* rc = plane + y0 * 256;
    const float* rp = plane + (y0 < 255 ? y0 + 1 : 255) * 256;
    const int xl = (x0 > 0) ? x0 - 1 : 0;
    // 3x3 sliding window registers (l,c,r columns x rows m,c,p)
    float m_l = rm[xl], m_c = rm[x0], m_r = rm[x0 + 1];
    float c_l = rc[xl], c_c = rc[x0], c_r = rc[x0 + 1];
    float p_l = rp[xl], p_c = rp[x0], p_r = rp[x0 + 1];
    for (int j = 0; j < 64; ++j) {
      const float v = c_c;
      const bool peak = !(m_l > v || m_c > v || m_r > v ||
                          c_l > v ||            c_r > v ||
                          p_l > v || p_c > v || p_r > v);
      if (peak) { if (j < 32) pb0 |= 1u << j; else pb1 |= 1u << (j - 32); }
      // shift window, load next column (x0+j+2, clamped)
      const int nx = (x0 + j + 2 < 256) ? x0 + j + 2 : 255;
      m_l = m_c; m_c = m_r; m_r = rm[nx];
      c_l = c_c; c_c = c_r; c_r = rc[nx];
      p_l = p_c; p_c = p_r; p_r = rp[nx];
    }
  }
  __syncthreads();

  // ---- 4-pass 8-bit radix select (descending) of top-100 peak keys ----
  unsigned prefix = 0u, mask = 0u;
  int krem = TOPK;
  for (int sh = 24; sh >= 0; sh -= 8) {
    if (tid < 256) hist[tid] = 0u;
    __syncthreads();
    // iterate only set bits of the peak mask
    for (unsigned w = 0; w < 2; ++w) {
      unsigned m = w ? pb1 : pb0;
      while (m) {
        const int j = (int)(w * 32) + __builtin_ctz(m);
        m &= m - 1;
        const unsigned key = fkey(plane[base + j]);
        if ((key & mask) == prefix) atomicAdd(&hist[(key >> sh) & 255u], 1u);
      }
    }
    __syncthreads();
    if (tid == 0) {
      unsigned acc = 0u; int d = 0;
      for (int bin = 255; bin >= 0; --bin) {
        const unsigned h = hist[bin];
        if (acc + h >= (unsigned)krem) { d = bin; break; }
        acc += h;
      }
      ctrl[0] = prefix | ((unsigned)d << sh);
      ctrl[1] = mask | (0xffu << sh);
      ctrl[2] = (unsigned)krem - acc;
    }
    __syncthreads();
    prefix = ctrl[0]; mask = ctrl[1]; krem = (int)ctrl[2];
  }

  // ---- collect exactly TOPK candidates into workspace ----
  unsigned long long* out_base = cand + (size_t)(b * CC + c) * TOPK;
  // distinct default fill (key 0 -> NaN score -> zero row downstream)
  if (tid < TOPK)
    out_base[tid] = ((unsigned long long)c << 16) | (unsigned)tid;
  if (tid == 0) { ctrl[3] = 0u; ctrl[4] = 0u; }
  __syncthreads();

  const unsigned T = prefix;
  const int G = TOPK - krem;   // count strictly greater than T
  for (unsigned w = 0; w < 2; ++w) {
    unsigned m = w ? pb1 : pb0;
    while (m) {
      const int j = (int)(w * 32) + __builtin_ctz(m);
      m &= m - 1;
      const unsigned key = fkey(plane[base + j]);
      if (key < T) continue;
      const unsigned long long rec =
          ((unsigned long long)key << 32) |
          (unsigned)((c << 16) | (base + j));
      if (key > T) {
        const unsigned p = atomicAdd(&ctrl[3], 1u);
        out_base[p] = rec;
      } else {
        const unsigned e = atomicAdd(&ctrl[4], 1u);
        if ((int)e < krem) out_base[G + e] = rec;
      }
    }
  }
}

// ---------------------------------------------------------------------------
// Kernel 2: one block per batch. Async-stage 8000 packed candidates into LDS,
// 8-pass u64 radix select of the distinct top-100, exact rank by pairwise
// compare among 100, decode + threshold + write det rows.
// LDS: 64000B cand + 1KB hist + 800B sel + ctrl.
// ---------------------------------------------------------------------------
__global__ __launch_bounds__(256) void k_final(
    const unsigned long long* __restrict__ cand,
    const float* __restrict__ wh, float* __restrict__ out) {
  const int b = blockIdx.x;
  const int tid = threadIdx.x;
  const int N = CC * TOPK;  // 8000

  extern __shared__ char smem[];
  unsigned long long* cl     = (unsigned long long*)smem;                 // 8000 u64
  unsigned*           hist   = (unsigned*)(smem + N * 8);                 // 256
  unsigned long long* sel    = (unsigned long long*)(smem + N * 8 + 1024);// 100 u64
  unsigned long long* ctrl64 = (unsigned long long*)(smem + N * 8 + 1024 + TOPK * 8);
  unsigned*           ctrl32 = (unsigned*)(ctrl64 + 2);                   // [0]=k [1]=gt [2]=eq

  const unsigned long long* src = cand + (size_t)b * N;
  for (int i = tid; i < N / 2; i += 256)
    async_copy16(src + (size_t)i * 2, cl + (size_t)i * 2);
  async_wait_all();
  __syncthreads();

  // ---- 8-pass u64 radix select: threshold T = 100th largest (all distinct) ----
  unsigned long long prefix = 0ull, mask = 0ull;
  int krem = TOPK;
  for (int sh = 56; sh >= 0; sh -= 8) {
    if (tid < 256) hist[tid] = 0u;
    __syncthreads();
    for (int i = tid; i < N; i += 256) {
      const unsigned long long v = cl[i];
      if ((v & mask) == prefix)
        atomicAdd(&hist[(unsigned)((v >> sh) & 255u)], 1u);
    }
    __syncthreads();
    if (tid == 0) {
      unsigned acc = 0u; int d = 0;
      for (int bin = 255; bin >= 0; --bin) {
        const unsigned h = hist[bin];
        if (acc + h >= (unsigned)krem) { d = bin; break; }
        acc += h;
      }
      ctrl64[0] = prefix | ((unsigned long long)d << sh);
      ctrl64[1] = mask | (255ull << sh);
      ctrl32[0] = (unsigned)krem - acc;
    }
    __syncthreads();
    prefix = ctrl64[0]; mask = ctrl64[1]; krem = (int)ctrl32[0];
  }

  // ---- collect top-100 into sel[] ----
  if (tid == 0) { ctrl32[1] = 0u; ctrl32[2] = 0u; }
  __syncthreads();
  const unsigned long long T = prefix;
  const int G = TOPK - krem;
  for (int i = tid; i < N; i += 256) {
    const unsigned long long v = cl[i];
    if (v > T) {
      const unsigned p = atomicAdd(&ctrl32[1], 1u);
      sel[p] = v;
    } else if (v == T) {
      const unsigned e = atomicAdd(&ctrl32[2], 1u);
      if ((int)e < krem) sel[G + e] = v;
    }
  }
  __syncthreads();

  // ---- exact descending rank among the 100 distinct winners; decode & emit ----
  if (tid < TOPK) {
    const unsigned long long my = sel[tid];
    int rank = 0;
    for (int q = 0; q < TOPK; ++q) rank += (sel[q] > my) ? 1 : 0;

    const unsigned key  = (unsigned)(my >> 32);
    const unsigned info = (unsigned)my;
    const int cls = (int)(info >> 16);
    const int s   = (int)(info & 0xffffu);

    const float v = finv(key);
    const float score = 1.0f / (1.0f + __expf(-v));  // NaN for invalid keys

    float* row = out + ((size_t)b * TOPK + rank) * 6;
    if (score > THR) {
      const float ys = (float)(s >> 8) * 4.0f;   // STRIDE = 4
      const float xs = (float)(s & 255) * 4.0f;
      const float* wb = wh + (size_t)b * 4 * HW;
      const float w0 = wb[0 * HW + s];
      const float w1 = wb[1 * HW + s];
      const float w2 = wb[2 * HW + s];
      const float w3 = wb[3 * HW + s];
      row[0] = xs - w0; row[1] = ys - w1;
      row[2] = xs + w2; row[3] = ys + w3;
      row[4] = score;   row[5] = (float)cls;
    } else {
      row[0] = 0.f; row[1] = 0.f; row[2] = 0.f;
      row[3] = 0.f; row[4] = 0.f; row[5] = 0.f;
    }
  }
}

// ---------------------------------------------------------------------------
extern "C" void kernel_launch(void* const* d_in, const int* in_sizes, int n_in,
                              void* d_out, int out_size, void* d_ws, size_t ws_size,
                              hipStream_t stream) {
  (void)in_sizes; (void)n_in; (void)out_size; (void)ws_size;
  const float* hm = (const float*)d_in[0];
  const float* wh = (const float*)d_in[1];
  float* out = (float*)d_out;
  unsigned long long* cand = (unsigned long long*)d_ws;  // 8*80*100 u64 = 512KB

  const size_t shm1 = (size_t)HW * 4 + 256 * 4 + 64;               // ~263KB (<320KB/WGP)
  k_class_topk<<<dim3(BB * CC), dim3(1024), shm1, stream>>>(hm, cand);

  const size_t shm2 = (size_t)CC * TOPK * 8 + 1024 + TOPK * 8 + 64; // ~66KB
  k_final<<<dim3(BB), dim3(256), shm2, stream>>>(cand, wh, out);
}